// LNCC_20237885899193
// MI455X (gfx1250) — compile-verified
//
#include <hip/hip_runtime.h>
#include <hip/hip_bf16.h>

typedef __attribute__((ext_vector_type(2))) float v2f;
typedef __attribute__((ext_vector_type(8))) float v8f;

#define DIM   160
#define TW    16     // tile width  (x)
#define THT   16     // tile height (y)
#define CZ    16     // z-chunk per block
#define WIN   3
#define KER   7
#define HALO  22     // 16 + 2*WIN
#define NVOX  8192000.0f   // 2*160^3
#define INV343 (1.0f / 343.0f)

// Deterministic wave32 reduction via V_WMMA_F32_16X16X4_F32.
// A-matrix (documented layout): lane L<16 -> row M=L, a[0]=K0, a[1]=K1;
//                               lane L>=16 -> row M=L-16, a[0]=K2, a[1]=K3.
// With B == all ones, D[m][n] = sum_k A[m][k] independent of B's lane layout.
// D (documented): lane L, reg j -> D[j + (L<16?0:8)][L&15].
// Full wave sum = (lane 0 value) + (lane 16 value).
__device__ __forceinline__ float wave_halfsum_wmma(float v) {
    v2f a; a[0] = v;    a[1] = 0.0f;
    v2f b; b[0] = 1.0f; b[1] = 1.0f;
    v8f c = {};
    v8f d = __builtin_amdgcn_wmma_f32_16x16x4_f32(
        /*neg_a=*/false, a, /*neg_b=*/false, b,
        /*c_mod=*/(short)0, c, /*reuse_a=*/false, /*reuse_b=*/false);
    return d[0] + d[1] + d[2] + d[3] + d[4] + d[5] + d[6] + d[7];
}

// Fast reciprocal: v_rcp_f32 + one Newton-Raphson step (<= 0.5 ulp).
// Denominator is >= 1e-6 so no special-case handling needed.
__device__ __forceinline__ float fast_rcp(float d) {
    float r = __builtin_amdgcn_rcpf(d);
    float e = __builtin_fmaf(-d, r, 1.0f);
    return __builtin_fmaf(r, e, r);
}

__global__ __launch_bounds__(256)
void lncc_fused_kernel(const float* __restrict__ I,
                       const float* __restrict__ J,
                       float* __restrict__ partial) {
    __shared__ float sI[HALO][24];          // raw halo slice, padded stride
    __shared__ float sJ[HALO][24];
    __shared__ float sX[5][HALO][TW];       // after x-pass (5 fields)
    __shared__ float red[16];

    const int tid = threadIdx.x;
    const int bx  = blockIdx.x;             // 0..9  (x tiles)
    const int by  = blockIdx.y;             // 0..9  (y tiles)
    const int bz  = blockIdx.z % 10;        // 0..9  (z chunks)
    const int bat = blockIdx.z / 10;        // 0..1  (batch)
    const int x0 = bx * TW, y0 = by * THT, z0 = bz * CZ;
    const size_t base = (size_t)bat * DIM * DIM * DIM;
    const int ox = tid & 15, oy = tid >> 4; // this thread's output (y,x)

    // Per-thread register shift buffer of z-window slices (static indexing only).
    float zb[5][KER];
#pragma unroll
    for (int f = 0; f < 5; ++f)
#pragma unroll
        for (int k = 0; k < KER; ++k) zb[f][k] = 0.0f;

    float acc = 0.0f;

    for (int zin = z0 - WIN; zin < z0 + CZ + WIN; ++zin) {
        __syncthreads();   // protect sI/sJ and sX from previous iteration's readers

        // ---- load raw halo slice (zero pad OOB) ----
        const bool zok = (zin >= 0) && (zin < DIM);
        for (int idx = tid; idx < HALO * HALO; idx += 256) {
            const int yy = idx / HALO;
            const int xx = idx - yy * HALO;
            const int gy = y0 - WIN + yy;
            const int gx = x0 - WIN + xx;
            float vi = 0.0f, vj = 0.0f;
            if (zok && (unsigned)gy < (unsigned)DIM && (unsigned)gx < (unsigned)DIM) {
                const size_t off = base + ((size_t)zin * DIM + gy) * DIM + gx;
                vi = I[off];
                vj = J[off];
                if ((xx == 0) && (zin + 1 < DIM)) {      // prefetch next z-slice row
                    __builtin_prefetch(&I[off + (size_t)DIM * DIM], 0, 3);
                    __builtin_prefetch(&J[off + (size_t)DIM * DIM], 0, 3);
                }
            }
            sI[yy][xx] = vi;
            sJ[yy][xx] = vj;
        }
        __syncthreads();

        // ---- x-pass: 7-tap sums of the 5 fields, halo rows ----
        for (int idx = tid; idx < HALO * TW; idx += 256) {
            const int y = idx >> 4;
            const int x = idx & 15;
            float si = 0.f, sj = 0.f, sii = 0.f, sjj = 0.f, sij = 0.f;
#pragma unroll
            for (int k = 0; k < KER; ++k) {
                const float a = sI[y][x + k];
                const float b = sJ[y][x + k];
                si  += a;     sj  += b;
                sii += a * a; sjj += b * b; sij += a * b;
            }
            sX[0][y][x] = si;  sX[1][y][x] = sj;
            sX[2][y][x] = sii; sX[3][y][x] = sjj; sX[4][y][x] = sij;
        }
        __syncthreads();

        // ---- y-pass, shifted into the per-thread register z-window ----
#pragma unroll
        for (int f = 0; f < 5; ++f) {
            float s = 0.f;
#pragma unroll
            for (int k = 0; k < KER; ++k) s += sX[f][oy + k][ox];
#pragma unroll
            for (int k = 0; k < KER - 1; ++k) zb[f][k] = zb[f][k + 1];
            zb[f][KER - 1] = s;
        }

        // ---- emit output once 7 z-slices are live (registers only) ----
        if (zin >= z0 + WIN) {
            float S[5];
#pragma unroll
            for (int f = 0; f < 5; ++f) {
                float s = 0.f;
#pragma unroll
                for (int k = 0; k < KER; ++k) s += zb[f][k];
                S[f] = s;
            }
            const float mI  = S[0] * INV343;
            const float mJ  = S[1] * INV343;
            const float mII = S[2] * INV343;
            const float mJJ = S[3] * INV343;
            const float mIJ = S[4] * INV343;
            const float cross = mIJ - mI * mJ;
            const float varI  = mII - mI * mI;
            const float varJ  = mJJ - mJ * mJ;
            acc = __builtin_fmaf(cross * cross, fast_rcp(varI * varJ + 1e-6f), acc);
        }
    }

    // ---- deterministic block reduction (WMMA wave reduce + fixed-order LDS) ----
    __syncthreads();                       // full block active; EXEC all-ones for WMMA
    const float h = wave_halfsum_wmma(acc);
    const int lane = tid & 31, w = tid >> 5;
    if (lane == 0)  red[2 * w]     = h;
    if (lane == 16) red[2 * w + 1] = h;
    __syncthreads();
    if (tid == 0) {
        float t = 0.f;
        for (int i = 0; i < 16; ++i) t += red[i];
        const int bid = blockIdx.x + gridDim.x * (blockIdx.y + gridDim.y * blockIdx.z);
        partial[bid] = t;
    }
}

__global__ __launch_bounds__(256)
void lncc_finalize_kernel(const float* __restrict__ partial, int n,
                          float* __restrict__ out) {
    __shared__ float red[16];
    const int tid = threadIdx.x;
    float acc = 0.f;
    for (int i = tid; i < n; i += 256) acc += partial[i];   // fixed order
    const float h = wave_halfsum_wmma(acc);
    const int lane = tid & 31, w = tid >> 5;
    if (lane == 0)  red[2 * w]     = h;
    if (lane == 16) red[2 * w + 1] = h;
    __syncthreads();
    if (tid == 0) {
        float t = 0.f;
        for (int i = 0; i < 16; ++i) t += red[i];
        out[0] = -t / NVOX;
    }
}

extern "C" void kernel_launch(void* const* d_in, const int* in_sizes, int n_in,
                              void* d_out, int out_size, void* d_ws, size_t ws_size,
                              hipStream_t stream) {
    const float* I = (const float*)d_in[0];
    const float* J = (const float*)d_in[1];
    float* out = (float*)d_out;
    float* ws  = (float*)d_ws;   // 2000 block partials (8 KB)

    dim3 grid(10, 10, 20);       // 10x10 xy tiles, 10 z-chunks * 2 batches
    lncc_fused_kernel<<<grid, 256, 0, stream>>>(I, J, ws);
    lncc_finalize_kernel<<<1, 256, 0, stream>>>(ws, 10 * 10 * 20, out);
}